// model003b_37666863186447
// MI455X (gfx1250) — compile-verified
//
#include <hip/hip_runtime.h>

// LRU (diagonal linear RNN), fully fused, segmented parallel scan.
//   NY=NU=8, NH=256 (128 rotation pairs), T=2048, NB=64.
// Pass 1: 16 segments x 64 batches local scans (zero init); every 16 steps the
//         X-chunk (16x256 f32) is projected to Y with V_WMMA_F32_16X16X4_F32,
//         split-K over the block's 4 waves. Segment-end states -> workspace.
// Pass 2: closed-form carry scan across segments (R^128 = r^128 * Rot(128*theta)).
// Pass 3: correction Y += W * R^{k+1} * c_s, same WMMA chunk projection.

typedef __attribute__((ext_vector_type(2))) float v2f;
typedef __attribute__((ext_vector_type(8))) float v8f;

#define NYv   8
#define NUv   8
#define NHv   256
#define H2v   128
#define T_LEN 2048
#define NBv   64
#define NSEG  16
#define SEG   (T_LEN / NSEG)   // 128
#define CHNK  16
#define NCH   (SEG / CHNK)     // 8
#define XPAD  258              // row stride (floats): even -> 8B-aligned b64 loads

// ---- B-operand for V_WMMA_F32_16X16X4_F32 -----------------------------------
// B is 4x16 (KxN) f32 in 2 VGPRs. Layout (mirrors documented A 16x4):
//   lanes 0-15 : N=lane,    v.x=K0 row, v.y=K1 row
//   lanes16-31 : N=lane-16, v.x=K2 row, v.y=K3 row
// Wt[k][n] = W_x2y[n][k] for n<8, else 0 (N padded 8->16).
__device__ __forceinline__ void load_wt_ops(const float* __restrict__ Wx2y,
                                            int wave, int lane, v2f bop[16]) {
  const int n  = lane & 15;
  const int kh = (lane >> 4) * 2;
#pragma unroll
  for (int i = 0; i < 16; ++i) {
    const int k0 = wave * 64 + 4 * i + kh;
    v2f v;
    if (n < NYv) {
      v.x = Wx2y[n * NHv + k0];
      v.y = Wx2y[n * NHv + k0 + 1];
    } else {
      v.x = 0.0f; v.y = 0.0f;
    }
    bop[i] = v;
  }
}

// ---- split-K WMMA projection of one 16x256 LDS chunk ------------------------
// Wave w accumulates K in [64w, 64w+64) with 16 WMMAs, writes 16x16 partial.
__device__ __forceinline__ void project_chunk(const float (*Xc)[XPAD],
                                              const v2f bop[16],
                                              int wave, int lane,
                                              float (*Yp)[CHNK][16]) {
  v8f acc = {};
  const int row = lane & 15;            // A: M = lane%16
  const int kh  = (lane >> 4) * 2;      // A: lanes>=16 hold K2,K3
#pragma unroll
  for (int i = 0; i < 16; ++i) {
    const int k0 = wave * 64 + 4 * i + kh;
    v2f a;
    a.x = Xc[row][k0];
    a.y = Xc[row][k0 + 1];
    acc = __builtin_amdgcn_wmma_f32_16x16x4_f32(
        false, a, false, bop[i], (short)0, acc, false, false);
  }
  // C layout: VGPR v -> M = v (lanes 0-15) or v+8 (lanes 16-31), N = lane%16
  const int mbase = (lane >> 4) * 8;
  const int ncol  = lane & 15;
#pragma unroll
  for (int v = 0; v < 8; ++v) Yp[wave][v + mbase][ncol] = acc[v];
}

// ---- Pass 1: local segment scans + chunked WMMA projection ------------------
__global__ void __launch_bounds__(128)
lru_seg_scan(const float* __restrict__ U,   const float* __restrict__ lr,
             const float* __restrict__ li,  const float* __restrict__ Bm,
             const float* __restrict__ mult,const float* __restrict__ Wx2y,
             const float* __restrict__ bx2y,float* __restrict__ Y,
             float* __restrict__ e_ws) {
  const int b = blockIdx.x;
  const int s = blockIdx.y;
  const int j = threadIdx.x;            // pair id 0..127
  const int wave = j >> 5, lane = j & 31;

  __shared__ float Uc[CHNK][NUv];
  __shared__ float Xc[CHNK][XPAD];
  __shared__ float Yp[4][CHNK][16];

  // Per-pair constants: scaled B rows, rotation (re, im).
  const float m1 = expf(mult[j]);
  const float m2 = expf(mult[j + H2v]);
  float Bs1[NUv], Bs2[NUv];
#pragma unroll
  for (int u = 0; u < NUv; ++u) {
    Bs1[u] = Bm[j * NUv + u] * m1;
    Bs2[u] = Bm[(j + H2v) * NUv + u] * m2;
  }
  const float r  = expf(-fabsf(lr[j]));
  const float th = 1.5707963267948966f * li[j];
  const float re = r * cosf(th);
  const float im = r * sinf(th);

  v2f bop[16];
  load_wt_ops(Wx2y, wave, lane, bop);

  float x1 = 0.0f, x2 = 0.0f;           // local state (zero init; carry in pass 3)

  // warm the cache for the first U chunk (global_prefetch_b8)
  if (j < CHNK)
    __builtin_prefetch(U + ((size_t)(s * SEG + j) * NBv + b) * NUv, 0, 0);

  for (int ch = 0; ch < NCH; ++ch) {
    const int t0 = s * SEG + ch * CHNK;
    if (j < 2 * CHNK) { // stage U chunk: 16 timesteps x 8 inputs, b128 loads
      const int tt = j >> 1, h = (j & 1) * 4;
      const float4 uv =
          *(const float4*)(U + ((size_t)(t0 + tt) * NBv + b) * NUv + h);
      Uc[tt][h + 0] = uv.x; Uc[tt][h + 1] = uv.y;
      Uc[tt][h + 2] = uv.z; Uc[tt][h + 3] = uv.w;
    }
    // prefetch next chunk's U rows while we scan/project this one
    if (ch + 1 < NCH && j < CHNK)
      __builtin_prefetch(U + ((size_t)(t0 + CHNK + j) * NBv + b) * NUv, 0, 0);
    __syncthreads();

#pragma unroll
    for (int tt = 0; tt < CHNK; ++tt) {
      float bu1 = 0.0f, bu2 = 0.0f;
#pragma unroll
      for (int u = 0; u < NUv; ++u) {
        const float uu = Uc[tt][u];
        bu1 = fmaf(uu, Bs1[u], bu1);
        bu2 = fmaf(uu, Bs2[u], bu2);
      }
      const float n1 = re * x1 - im * x2 + bu1;
      const float n2 = im * x1 + re * x2 + bu2;
      x1 = n1; x2 = n2;
      Xc[tt][j]       = x1;
      Xc[tt][j + H2v] = x2;
    }
    __syncthreads();

    project_chunk(Xc, bop, wave, lane, Yp);
    __syncthreads();

    { // reduce 4 wave partials + bias, write Y
      const int tt = j >> 3, n = j & 7;
      const float y = Yp[0][tt][n] + Yp[1][tt][n] + Yp[2][tt][n] + Yp[3][tt][n]
                    + bx2y[n];
      Y[((size_t)(t0 + tt) * NBv + b) * NYv + n] = y;
    }
    __syncthreads();
  }

  // segment-end local state -> workspace
  e_ws[((size_t)s * NBv + b) * NHv + j]       = x1;
  e_ws[((size_t)s * NBv + b) * NHv + j + H2v] = x2;
}

// ---- Pass 2: carry scan across segments (closed-form R^SEG) -----------------
__global__ void __launch_bounds__(128)
lru_carry(const float* __restrict__ y0,  const float* __restrict__ Wy2x,
          const float* __restrict__ by2x,const float* __restrict__ lr,
          const float* __restrict__ li,  const float* __restrict__ e_ws,
          float* __restrict__ c_ws) {
  const int b = blockIdx.x;
  const int j = threadIdx.x;            // pair id

  const float th  = 1.5707963267948966f * li[j];
  const float rS  = expf(-(float)SEG * fabsf(lr[j]));   // r^SEG
  const float phS = (float)SEG * th;
  const float reS = rS * cosf(phS);
  const float imS = rS * sinf(phS);

  // x0 = y0 @ W_y2x^T + b_y2x
  float c1 = by2x[j], c2 = by2x[j + H2v];
#pragma unroll
  for (int n = 0; n < NYv; ++n) {
    const float yv = y0[b * NYv + n];
    c1 = fmaf(yv, Wy2x[j * NYv + n], c1);
    c2 = fmaf(yv, Wy2x[(j + H2v) * NYv + n], c2);
  }

  for (int s = 0; s < NSEG; ++s) {
    const size_t base = ((size_t)s * NBv + b) * NHv;
    c_ws[base + j]       = c1;
    c_ws[base + j + H2v] = c2;
    const float e1 = e_ws[base + j];
    const float e2 = e_ws[base + j + H2v];
    const float n1 = reS * c1 - imS * c2 + e1;
    const float n2 = imS * c1 + reS * c2 + e2;
    c1 = n1; c2 = n2;
  }
}

// ---- Pass 3: Y += W * R^{k+1} * c_s (chunked WMMA, RMW on Y) ----------------
__global__ void __launch_bounds__(128)
lru_correct(const float* __restrict__ lr, const float* __restrict__ li,
            const float* __restrict__ Wx2y, const float* __restrict__ c_ws,
            float* __restrict__ Y) {
  const int b = blockIdx.x;
  const int s = blockIdx.y;
  const int j = threadIdx.x;
  const int wave = j >> 5, lane = j & 31;

  __shared__ float Xc[CHNK][XPAD];
  __shared__ float Yp[4][CHNK][16];

  const float r  = expf(-fabsf(lr[j]));
  const float th = 1.5707963267948966f * li[j];
  const float re = r * cosf(th);
  const float im = r * sinf(th);

  v2f bop[16];
  load_wt_ops(Wx2y, wave, lane, bop);

  const size_t cbase = ((size_t)s * NBv + b) * NHv;
  float v1 = c_ws[cbase + j];
  float v2 = c_ws[cbase + j + H2v];

  for (int ch = 0; ch < NCH; ++ch) {
    const int t0 = s * SEG + ch * CHNK;
#pragma unroll
    for (int tt = 0; tt < CHNK; ++tt) {
      const float n1 = re * v1 - im * v2;   // v <- R v  (gives R^{k+1} c)
      const float n2 = im * v1 + re * v2;
      v1 = n1; v2 = n2;
      Xc[tt][j]       = v1;
      Xc[tt][j + H2v] = v2;
    }
    __syncthreads();

    project_chunk(Xc, bop, wave, lane, Yp);
    __syncthreads();

    {
      const int tt = j >> 3, n = j & 7;
      const size_t yi = ((size_t)(t0 + tt) * NBv + b) * NYv + n;
      Y[yi] += Yp[0][tt][n] + Yp[1][tt][n] + Yp[2][tt][n] + Yp[3][tt][n];
    }
    __syncthreads();
  }
}

extern "C" void kernel_launch(void* const* d_in, const int* in_sizes, int n_in,
                              void* d_out, int out_size, void* d_ws, size_t ws_size,
                              hipStream_t stream) {
  (void)in_sizes; (void)n_in; (void)out_size; (void)ws_size;
  const float* y0   = (const float*)d_in[0];
  const float* U    = (const float*)d_in[1];
  const float* lr   = (const float*)d_in[2];
  const float* li   = (const float*)d_in[3];
  const float* Bm   = (const float*)d_in[4];
  const float* mult = (const float*)d_in[5];
  const float* Wy2x = (const float*)d_in[6];
  const float* by2x = (const float*)d_in[7];
  const float* Wx2y = (const float*)d_in[8];
  const float* bx2y = (const float*)d_in[9];
  float* Y = (float*)d_out;

  float* e_ws = (float*)d_ws;                          // NSEG*NB*NH f32 (1 MB)
  float* c_ws = e_ws + (size_t)NSEG * NBv * NHv;       // NSEG*NB*NH f32 (1 MB)

  dim3 grid(NBv, NSEG);
  lru_seg_scan<<<grid, 128, 0, stream>>>(U, lr, li, Bm, mult, Wx2y, bx2y, Y, e_ws);
  lru_carry<<<NBv, 128, 0, stream>>>(y0, Wy2x, by2x, lr, li, e_ws, c_ws);
  lru_correct<<<grid, 128, 0, stream>>>(lr, li, Wx2y, c_ws, Y);
}